// SSM_56195352101081
// MI455X (gfx1250) — compile-verified
//
#include <hip/hip_runtime.h>
#include <stdint.h>

// Problem constants (fixed by the reference: b=2, s=2048, d=1024, n=16)
#define BATCH 2
#define SEQ   2048
#define DDIM  1024
#define NDIM  16

#define DTILE  4                  // d-values per block
#define BLOCK  64                 // 4 d x 16 n threads = 2 waves (wave32)
#define TSTEP  16                 // timesteps staged per pipeline stage
#define NBUF   6                  // LDS stage buffers (pipeline depth NBUF-1)
#define NSTAGES (SEQ / TSTEP)     // 128

#define AB_STEP_BYTES (DDIM * NDIM * 4)   // 65536: step stride of A/B in global
#define X_STEP_BYTES  (DDIM * 4)          // 4096 : step stride of x in global
#define AB_TILE_FLOATS (DTILE * NDIM)     // 64 contiguous floats per step (256B)

static_assert(NSTAGES * TSTEP == SEQ, "tiling");
static_assert(2 * NBUF <= 63, "TENSORcnt budget");
static_assert(2 * NBUF <= 63, "ASYNCcnt budget");

typedef unsigned int v4u __attribute__((ext_vector_type(4)));
typedef int          v8i __attribute__((ext_vector_type(8)));
typedef int          v4i __attribute__((ext_vector_type(4)));

// ---- CDNA5 async global->LDS copies (ASYNCcnt-tracked) ---------------------
__device__ __forceinline__ void async_ld_b128(uint32_t lds_off, const void* gaddr) {
    asm volatile("global_load_async_to_lds_b128 %0, %1, off"
                 :: "v"(lds_off), "v"(gaddr) : "memory");
}
__device__ __forceinline__ void async_ld_b32(uint32_t lds_off, const void* gaddr) {
    asm volatile("global_load_async_to_lds_b32 %0, %1, off"
                 :: "v"(lds_off), "v"(gaddr) : "memory");
}
__device__ __forceinline__ uint32_t lds_off32(const void* p) {
    // generic pointer to __shared__ : low 32 bits are the LDS byte offset
    return (uint32_t)(uintptr_t)p;
}

// ---- CDNA5 Tensor Data Mover: 2D tile (16 rows x 256B, row stride 64KB) ----
// D# per CDNA5 ISA; dims == tile dims => no OOB clipping.
__device__ __forceinline__ void tdm_load_tile_16x256B(uint32_t lds_byte_off,
                                                      const void* gaddr) {
    const uint64_t ga = (uint64_t)(uintptr_t)gaddr;
    v4u g0;
    g0.x = 1u;                                   // count=1 (valid), user mode
    g0.y = lds_byte_off;                         // lds_addr
    g0.z = (uint32_t)ga;                         // global_addr[31:0]
    g0.w = (uint32_t)(ga >> 32) | 0x80000000u;   // global_addr[56:32] | type=2
    v8i g1;
    g1.s0 = 0x00020000;                          // wg_mask=0, data_size=2 (4B)
    g1.s1 = (int)(64u << 16);                    // tensor_dim0 = 64 elements
    g1.s2 = (int)(16u << 16);                    // tensor_dim1 = 16 rows
    g1.s3 = (int)(64u << 16);                    // tile_dim0   = 64 elements
    g1.s4 = 16;                                  // tile_dim1=16, tile_dim2=0 (2D)
    g1.s5 = DDIM * NDIM;                         // tensor_dim0_stride = 16384
    g1.s6 = 0;
    g1.s7 = 0;
    const v4i z4 = {0, 0, 0, 0};
    const v8i z8 = {0, 0, 0, 0, 0, 0, 0, 0};
    __builtin_amdgcn_tensor_load_to_lds(g0, g1, z4, z4, z8, 0);
}

// ---- 16-lane sum via DPP (no LDS traffic): xor1,xor2,xor4 (DPP8) + row_mirror
#define DPP8SEL(a,b,c,d,e,f,g,h) \
    ((a)|((b)<<3)|((c)<<6)|((d)<<9)|((e)<<12)|((f)<<15)|((g)<<18)|((h)<<21))
#define DPP_ROW_MIRROR 0x140

__device__ __forceinline__ float sum16_dpp(float p) {
    int v = __float_as_int(p);
    p += __int_as_float(__builtin_amdgcn_mov_dpp8(v, DPP8SEL(1,0,3,2,5,4,7,6)));   // xor1
    v = __float_as_int(p);
    p += __int_as_float(__builtin_amdgcn_mov_dpp8(v, DPP8SEL(2,3,0,1,6,7,4,5)));   // xor2
    v = __float_as_int(p);
    p += __int_as_float(__builtin_amdgcn_mov_dpp8(v, DPP8SEL(4,5,6,7,0,1,2,3)));   // xor4
    v = __float_as_int(p);
    p += __int_as_float(__builtin_amdgcn_update_dpp(0, v, DPP_ROW_MIRROR, 0xf, 0xf, true)); // xor8
    return p;
}

__global__ __launch_bounds__(BLOCK) void ssm_scan_kernel(
    const float* __restrict__ A,  // [b, s, d, n]
    const float* __restrict__ B,  // [b, s, d, n]
    const float* __restrict__ C,  // [b, s, n]
    const float* __restrict__ X,  // [b, s, d]
    float* __restrict__ Y)        // [b, s, d]
{
    __shared__ float sA[NBUF][TSTEP * AB_TILE_FLOATS];  // per stage: [ts][di*16+ni]
    __shared__ float sB[NBUF][TSTEP * AB_TILE_FLOATS];
    __shared__ float sX[NBUF][TSTEP * DTILE];           // per stage: [ts][di]
    __shared__ float sC[NBUF][TSTEP * NDIM];            // per stage: [ts][ni]

    const int tid = threadIdx.x;      // 0..63
    const int di  = tid >> 4;         // 0..3
    const int ni  = tid & 15;         // 0..15
    const int bi  = blockIdx.x >> 8;              // 2 batches x 256 d-tiles
    const int d0  = (blockIdx.x & 255) * DTILE;

    // Global byte bases for this block's tile at t=0
    const char* Ab = (const char*)(A + ((size_t)bi * SEQ * DDIM + d0) * NDIM);
    const char* Bb = (const char*)(B + ((size_t)bi * SEQ * DDIM + d0) * NDIM);
    const char* Xb = (const char*)(X + (size_t)bi * SEQ * DDIM + d0);
    const char* Cb = (const char*)(C + (size_t)bi * SEQ * NDIM);

    auto prefetch = [&](int k) {
        const int buf = k % NBUF;
        const size_t t0 = (size_t)k * TSTEP;
        // A and B tiles: one TDM descriptor each; wave 0 only (TDM ignores EXEC,
        // so the second wave must branch around the instruction entirely).
        if (tid < 32) {
            tdm_load_tile_16x256B(lds_off32(&sA[buf][0]), Ab + t0 * AB_STEP_BYTES);
            tdm_load_tile_16x256B(lds_off32(&sB[buf][0]), Bb + t0 * AB_STEP_BYTES);
        }
        // x tile: 16 steps x 16B (1 x B32, all 64 threads)
        {
            const int ts = tid >> 2, e = tid & 3;
            const char* g = Xb + (t0 + ts) * X_STEP_BYTES + (size_t)e * 4;
            async_ld_b32(lds_off32(&sX[buf][ts * DTILE + e]), g);
        }
        // C tile: 256 contiguous floats (1 x B128)
        {
            const char* g = Cb + t0 * NDIM * 4 + (size_t)tid * 16;
            async_ld_b128(lds_off32(&sC[buf][0]) + tid * 16, g);
        }
    };

    float h = 0.0f;  // recurrence state for (bi, d0+di, ni)
    const size_t ybase = (size_t)bi * SEQ * DDIM + d0 + di;

    // Fill the pipeline: stages 0 .. NBUF-2
#pragma unroll
    for (int k = 0; k < NBUF - 1; ++k) prefetch(k);

    for (int k = 0; k < NSTAGES; ++k) {
        if (k + NBUF - 1 < NSTAGES) {
            prefetch(k + NBUF - 1);
            // 6 stages outstanding (12 TDM / 12 async); <=10 => oldest landed
            __builtin_amdgcn_s_wait_tensorcnt(2 * (NBUF - 1));
            asm volatile("s_wait_asynccnt 0xa" ::: "memory");
        } else {
            __builtin_amdgcn_s_wait_tensorcnt(0);
            asm volatile("s_wait_asynccnt 0x0" ::: "memory");  // drain
        }
        __syncthreads();  // all waves' stage-k DMA visible; prior compute done

        const int buf = k % NBUF;
#pragma unroll
        for (int ts = 0; ts < TSTEP; ++ts) {
            float a  = sA[buf][ts * AB_TILE_FLOATS + tid];
            float bb = sB[buf][ts * AB_TILE_FLOATS + tid];
            float xv = sX[buf][ts * DTILE + di];
            float cv = sC[buf][ts * NDIM + ni];
            h = fmaf(a, h, bb * xv);          // h_t = A_t*h + B_t*x_t
            float p = sum16_dpp(cv * h);      // contract over n (16 lanes)
            if (ni == 0)
                Y[ybase + (size_t)(k * TSTEP + ts) * DDIM] = p;
        }
        __syncthreads();  // buffer free before it is DMA-overwritten next iter
    }
}

extern "C" void kernel_launch(void* const* d_in, const int* in_sizes, int n_in,
                              void* d_out, int out_size, void* d_ws, size_t ws_size,
                              hipStream_t stream) {
    (void)in_sizes; (void)n_in; (void)out_size; (void)d_ws; (void)ws_size;
    const float* A = (const float*)d_in[0];
    const float* B = (const float*)d_in[1];
    const float* C = (const float*)d_in[2];
    const float* X = (const float*)d_in[3];
    float* Y = (float*)d_out;

    dim3 grid(BATCH * (DDIM / DTILE));  // 512 blocks
    dim3 block(BLOCK);                  // 64 threads = 2 waves (wave32)
    ssm_scan_kernel<<<grid, block, 0, stream>>>(A, B, C, X, Y);
}